// TokenMixer_41394894799710
// MI455X (gfx1250) — compile-verified
//
#include <hip/hip_runtime.h>
#include <math.h>

// ---------------- problem constants (from reference) ----------------
#define BB   4
#define CC   128
#define HH   256
#define WWD  256
#define WF   129          // rfft bins along W
#define NB   8            // AFNO blocks
#define BS   16           // block size (channels per block)
#define HWP  (HH*WWD)     // 65536 pixels per (b,c) plane
#define CHW  (CC*HWP)     // per-batch stride of [B,C,H,W]
#define CHF  (HH*WF)      // per-channel stride of spectral plane = 33024
#define MTILES 2064       // 33024/16 mode-tiles per batch
#define LAMBDA 0.01f      // softshrink
#define PI_F 3.14159265358979323846f

typedef __attribute__((ext_vector_type(2))) float v2f;
typedef __attribute__((ext_vector_type(8))) float v8f;

// CDNA5 fp32 WMMA: D(16x16) += A(16x4) * B(4x16), exact fp32 (matches reference).
__device__ __forceinline__ v8f wmma_f32_16x16x4(v2f a, v2f b, v8f c) {
  return __builtin_amdgcn_wmma_f32_16x16x4_f32(false, a, false, b, (short)0, c,
                                               false, false);
}

// =====================================================================
// Kernel 1/7: 1x1 conv GEMM. D[M=out_ch][N=pixel] = W[M][K] * X[K][N].
// One wave computes a 16x16 tile; K=128 in 32 WMMA steps.
// EPI==0: +bias, exact GELU.  EPI==1: +bias, +residual (global shortcut).
// A 16x4 f32 layout: lanes 0-15 M, VGPR0/1 = K {0,1}; lanes 16-31 K {2,3}.
// B 4x16 f32 layout: lanes 0-15 N with K rows {0,1}; lanes 16-31 K rows {2,3}.
// C/D layout: VGPR r -> M = r (+8 for upper lane half), N = lane&15.
// =====================================================================
template<int EPI>
__global__ void __launch_bounds__(256)
conv1x1_wmma(const float* __restrict__ X, const float* __restrict__ Wm,
             const float* __restrict__ bias, const float* __restrict__ res,
             float* __restrict__ Y) {
  const int lane = threadIdx.x & 31;
  const int w    = blockIdx.x * 8 + (threadIdx.x >> 5);
  const int nt   = w & 4095;          // 4096 pixel tiles of 16
  const int mt   = (w >> 12) & 7;     // 8 out-channel tiles
  const int b    = w >> 15;           // batch
  const int nn   = lane & 15;         // A:M index / B,D:N index
  const int half = lane >> 4;
  const int kb0  = half * 2;

  const float* Xb   = X  + (size_t)b * CHW + nt * 16;
  const float* Arow = Wm + (size_t)(mt * 16 + nn) * CC;

  v8f acc = {0.f, 0.f, 0.f, 0.f, 0.f, 0.f, 0.f, 0.f};
#pragma unroll
  for (int k = 0; k < CC; k += 4) {
    v2f a, bf;
    a.x  = Arow[k + kb0];
    a.y  = Arow[k + kb0 + 1];
    bf.x = Xb[(size_t)(k + kb0) * HWP + nn];
    bf.y = Xb[(size_t)(k + kb0 + 1) * HWP + nn];
    acc = wmma_f32_16x16x4(a, bf, acc);
  }

  float* Yb = Y + (size_t)b * CHW + (size_t)(mt * 16) * HWP + nt * 16;
#pragma unroll
  for (int r = 0; r < 8; ++r) {
    const int o = r + half * 8;
    float v = acc[r] + bias[mt * 16 + o];
    if (EPI == 0) {
      v = 0.5f * v * (1.0f + erff(v * 0.70710678118654752f));  // exact GELU
    } else {
      v += res[(size_t)b * CHW + (size_t)(mt * 16 + o) * HWP + nt * 16 + nn];
    }
    Yb[(size_t)o * HWP + nn] = v;
  }
}

// =====================================================================
// Kernel 2/7: forward rfft over W (real 256 -> 129 complex), ortho 1/16.
// Stockham radix-2 autosort in LDS; 2 rows per 256-thread block.
// =====================================================================
__global__ void __launch_bounds__(256)
fft_rows_fwd(const float* __restrict__ in, float* __restrict__ sr,
             float* __restrict__ si) {
  __shared__ float2 lds[2][2][256];
  const int rl  = threadIdx.x >> 7;
  const int j   = threadIdx.x & 127;
  const int row = blockIdx.x * 2 + rl;           // (b,c,h) flattened
  const float* src = in + (size_t)row * WWD;
  lds[rl][0][j]       = make_float2(src[j], 0.f);
  lds[rl][0][j + 128] = make_float2(src[j + 128], 0.f);
  __syncthreads();
  int pp = 0;
#pragma unroll
  for (int Ns = 1; Ns < 256; Ns <<= 1) {
    const float2 v0 = lds[rl][pp][j];
    const float2 v1 = lds[rl][pp][j + 128];
    const int jm = j & (Ns - 1);
    const float ang = -PI_F * (float)jm / (float)Ns;
    float s, c;
    __sincosf(ang, &s, &c);
    const float tr = v1.x * c - v1.y * s;
    const float ti = v1.x * s + v1.y * c;
    const int e = ((j & ~(Ns - 1)) << 1) | jm;
    lds[rl][pp ^ 1][e]      = make_float2(v0.x + tr, v0.y + ti);
    lds[rl][pp ^ 1][e + Ns] = make_float2(v0.x - tr, v0.y - ti);
    pp ^= 1;
    __syncthreads();
  }
  const size_t ob = (size_t)row * WF;
  const float2 v = lds[rl][0][j];
  sr[ob + j] = v.x * 0.0625f;
  si[ob + j] = v.y * 0.0625f;
  if (j == 0) {
    const float2 vn = lds[rl][0][128];
    sr[ob + 128] = vn.x * 0.0625f;
    si[ob + 128] = vn.y * 0.0625f;
  }
}

// =====================================================================
// Kernels 3/7 & 5/7: complex FFT over H (SIGN=-1 fwd, +1 inv), ortho 1/16.
// 8 adjacent wf columns per block for coalesced global access.
// =====================================================================
template<int SIGN>
__global__ void __launch_bounds__(256)
fft_cols(float* __restrict__ sr, float* __restrict__ si) {
  __shared__ float2 lds[2][8][256];
  const int tid = threadIdx.x;
  const int grp = blockIdx.x % 17;       // 17 groups of 8 cover WF=129
  const int bc  = blockIdx.x / 17;       // (b*C + c)
  const int wf0 = grp * 8;
  const size_t base = (size_t)bc * CHF;
  for (int u = tid; u < 2048; u += 256) {
    const int hh = u >> 3, c8 = u & 7;
    const int wf = wf0 + c8;
    float re = 0.f, im = 0.f;
    if (wf < WF) {
      re = sr[base + (size_t)hh * WF + wf];
      im = si[base + (size_t)hh * WF + wf];
    }
    lds[0][c8][hh] = make_float2(re, im);
  }
  __syncthreads();
  int pp = 0;
  for (int Ns = 1; Ns < 256; Ns <<= 1) {
    for (int u = tid; u < 1024; u += 256) {
      const int c8 = u >> 7, j = u & 127;
      const float2 v0 = lds[pp][c8][j];
      const float2 v1 = lds[pp][c8][j + 128];
      const int jm = j & (Ns - 1);
      const float ang = (float)SIGN * PI_F * (float)jm / (float)Ns;
      float s, c;
      __sincosf(ang, &s, &c);
      const float tr = v1.x * c - v1.y * s;
      const float ti = v1.x * s + v1.y * c;
      const int e = ((j & ~(Ns - 1)) << 1) | jm;
      lds[pp ^ 1][c8][e]      = make_float2(v0.x + tr, v0.y + ti);
      lds[pp ^ 1][c8][e + Ns] = make_float2(v0.x - tr, v0.y - ti);
    }
    __syncthreads();
    pp ^= 1;
  }
  for (int u = tid; u < 2048; u += 256) {
    const int hh = u >> 3, c8 = u & 7;
    const int wf = wf0 + c8;
    if (wf < WF) {
      const float2 v = lds[pp][c8][hh];   // pp==0 after 8 stages
      sr[base + (size_t)hh * WF + wf] = v.x * 0.0625f;
      si[base + (size_t)hh * WF + wf] = v.y * 0.0625f;
    }
  }
}

// =====================================================================
// Kernel 4/7: block-diagonal complex MLP in Fourier space, in-place.
// Computed transposed: D[o][mode] = sum_k W[k][o] * X[k][mode], so the
// mode axis sits on lanes (contiguous in memory -> coalesced).
// One wave = 16 modes x 1 block. 32 v_wmma_f32_16x16x4_f32 per wave.
// A/B NEG is unavailable for f32 WMMA (ISA: only CNeg), so the complex
// "minus" terms negate the (tiny) weight A-fragment in VALU.
// Layer1 -> Layer2 relayout (C/D layout -> B layout) via per-wave LDS tile.
// =====================================================================
__global__ void __launch_bounds__(256)
block_mlp_wmma(float* __restrict__ sr, float* __restrict__ si,
               const float* __restrict__ w1, const float* __restrict__ b1,
               const float* __restrict__ w2, const float* __restrict__ b2) {
  __shared__ float o1r_t[8][256];
  __shared__ float o1i_t[8][256];
  const int lane = threadIdx.x & 31;
  const int wv   = threadIdx.x >> 5;
  const int gw   = blockIdx.x * 8 + wv;       // wave-tile id
  const int n    = gw & 7;                    // AFNO block
  const int t    = (gw >> 3) % MTILES;        // mode tile within batch
  const int b    = (gw >> 3) / MTILES;        // batch
  const int o    = lane & 15;                 // A:M(out-ch) / B,D:N(mode)
  const int half = lane >> 4;
  const int kb   = half * 2;
  const int mm   = t * 16 + o;                // flat (h*Wf+wf) mode index
  const size_t sb = (size_t)b * CC * CHF + (size_t)(n * 16) * CHF;

  const float* W1r = w1 + n * 256;
  const float* W1i = w1 + NB * 256 + n * 256;
  const float* W2r = w2 + n * 256;
  const float* W2i = w2 + NB * 256 + n * 256;

  // ---- layer 1: o1 = relu(X * W1 + b1), complex, accumulate bias in C ----
  v8f ar, ai;
#pragma unroll
  for (int r = 0; r < 8; ++r) {
    ar[r] = b1[n * 16 + r + half * 8];
    ai[r] = b1[NB * BS + n * 16 + r + half * 8];
  }
#pragma unroll
  for (int s4 = 0; s4 < 16; s4 += 4) {
    v2f a1, a2, a2n, br, bi;
    a1.x = W1r[(s4 + kb) * 16 + o];  a1.y = W1r[(s4 + kb + 1) * 16 + o];
    a2.x = W1i[(s4 + kb) * 16 + o];  a2.y = W1i[(s4 + kb + 1) * 16 + o];
    br.x = sr[sb + (size_t)(s4 + kb) * CHF + mm];
    br.y = sr[sb + (size_t)(s4 + kb + 1) * CHF + mm];
    bi.x = si[sb + (size_t)(s4 + kb) * CHF + mm];
    bi.y = si[sb + (size_t)(s4 + kb + 1) * CHF + mm];
    a2n.x = -a2.x; a2n.y = -a2.y;
    ar = wmma_f32_16x16x4(a1,  br, ar);   // + W1r^T * xr
    ar = wmma_f32_16x16x4(a2n, bi, ar);   // - W1i^T * xi
    ai = wmma_f32_16x16x4(a1,  bi, ai);   // + W1r^T * xi
    ai = wmma_f32_16x16x4(a2,  br, ai);   // + W1i^T * xr
  }
  // ReLU, stash in LDS [i][mode_local] for layer-2 B fragments
#pragma unroll
  for (int r = 0; r < 8; ++r) {
    o1r_t[wv][(r + half * 8) * 16 + o] = ar[r] > 0.f ? ar[r] : 0.f;
    o1i_t[wv][(r + half * 8) * 16 + o] = ai[r] > 0.f ? ai[r] : 0.f;
  }
  __syncthreads();

  // ---- layer 2: o2 = softshrink(o1 * W2 + b2), complex ----
  v8f cr, ci;
#pragma unroll
  for (int r = 0; r < 8; ++r) {
    cr[r] = b2[n * 16 + r + half * 8];
    ci[r] = b2[NB * BS + n * 16 + r + half * 8];
  }
#pragma unroll
  for (int s4 = 0; s4 < 16; s4 += 4) {
    v2f a1, a2, a2n, br, bi;
    a1.x = W2r[(s4 + kb) * 16 + o];  a1.y = W2r[(s4 + kb + 1) * 16 + o];
    a2.x = W2i[(s4 + kb) * 16 + o];  a2.y = W2i[(s4 + kb + 1) * 16 + o];
    br.x = o1r_t[wv][(s4 + kb) * 16 + o];
    br.y = o1r_t[wv][(s4 + kb + 1) * 16 + o];
    bi.x = o1i_t[wv][(s4 + kb) * 16 + o];
    bi.y = o1i_t[wv][(s4 + kb + 1) * 16 + o];
    a2n.x = -a2.x; a2n.y = -a2.y;
    cr = wmma_f32_16x16x4(a1,  br, cr);
    cr = wmma_f32_16x16x4(a2n, bi, cr);
    ci = wmma_f32_16x16x4(a1,  bi, ci);
    ci = wmma_f32_16x16x4(a2,  br, ci);
  }
#pragma unroll
  for (int r = 0; r < 8; ++r) {
    const int oc = r + half * 8;
    float vr = cr[r], vi = ci[r];
    vr = (vr > LAMBDA) ? vr - LAMBDA : ((vr < -LAMBDA) ? vr + LAMBDA : 0.f);
    vi = (vi > LAMBDA) ? vi - LAMBDA : ((vi < -LAMBDA) ? vi + LAMBDA : 0.f);
    sr[sb + (size_t)oc * CHF + mm] = vr;
    si[sb + (size_t)oc * CHF + mm] = vi;
  }
}

// =====================================================================
// Kernel 6/7: inverse rfft over W + AFNO internal residual (in-place on h1).
// Rebuild full 256-bin spectrum via Hermitian symmetry, inverse complex
// FFT, take real part (== irfft), ortho 1/16, add residual.
// =====================================================================
__global__ void __launch_bounds__(256)
irfft_rows(const float* __restrict__ sr, const float* __restrict__ si,
           float* __restrict__ h) {
  __shared__ float2 lds[2][2][256];
  const int rl  = threadIdx.x >> 7;
  const int j   = threadIdx.x & 127;
  const int row = blockIdx.x * 2 + rl;
  const size_t ib = (size_t)row * WF;
  lds[rl][0][j] = make_float2(sr[ib + j], si[ib + j]);
  if (j == 0) {
    lds[rl][0][128] = make_float2(sr[ib + 128], si[ib + 128]);
  } else {
    const int k2 = 128 - j;                          // conj mirror
    lds[rl][0][j + 128] = make_float2(sr[ib + k2], -si[ib + k2]);
  }
  __syncthreads();
  int pp = 0;
#pragma unroll
  for (int Ns = 1; Ns < 256; Ns <<= 1) {
    const float2 v0 = lds[rl][pp][j];
    const float2 v1 = lds[rl][pp][j + 128];
    const int jm = j & (Ns - 1);
    const float ang = PI_F * (float)jm / (float)Ns;  // inverse: +
    float s, c;
    __sincosf(ang, &s, &c);
    const float tr = v1.x * c - v1.y * s;
    const float ti = v1.x * s + v1.y * c;
    const int e = ((j & ~(Ns - 1)) << 1) | jm;
    lds[rl][pp ^ 1][e]      = make_float2(v0.x + tr, v0.y + ti);
    lds[rl][pp ^ 1][e + Ns] = make_float2(v0.x - tr, v0.y - ti);
    pp ^= 1;
    __syncthreads();
  }
  float* dst = h + (size_t)row * WWD;
  dst[j]       = lds[rl][0][j].x       * 0.0625f + dst[j];
  dst[j + 128] = lds[rl][0][j + 128].x * 0.0625f + dst[j + 128];
}

// =====================================================================
extern "C" void kernel_launch(void* const* d_in, const int* in_sizes, int n_in,
                              void* d_out, int out_size, void* d_ws,
                              size_t ws_size, hipStream_t stream) {
  (void)in_sizes; (void)n_in; (void)out_size; (void)ws_size;
  const float* x   = (const float*)d_in[0];
  const float* p1w = (const float*)d_in[1];
  const float* p1b = (const float*)d_in[2];
  const float* w1  = (const float*)d_in[3];
  const float* b1  = (const float*)d_in[4];
  const float* w2  = (const float*)d_in[5];
  const float* b2  = (const float*)d_in[6];
  const float* p2w = (const float*)d_in[7];
  const float* p2b = (const float*)d_in[8];
  float* out = (float*)d_out;

  float* h1 = (float*)d_ws;                                 // [B,C,H,W]   128 MB
  float* sr = h1 + (size_t)BB * CHW;                        // [B,C,H,WF]  64.5 MB
  float* si = sr + (size_t)BB * CC * CHF;                   // [B,C,H,WF]  64.5 MB

  // 1) h1 = gelu(p1_w @ x + p1_b)
  conv1x1_wmma<0><<<16384, 256, 0, stream>>>(x, p1w, p1b, nullptr, h1);
  // 2) rfft over W (ortho 1/sqrt(W))
  fft_rows_fwd<<<BB * CC * HH / 2, 256, 0, stream>>>(h1, sr, si);
  // 3) FFT over H (ortho 1/sqrt(H))
  fft_cols<-1><<<BB * CC * 17, 256, 0, stream>>>(sr, si);
  // 4) block-diagonal complex MLP + softshrink, in-place
  block_mlp_wmma<<<BB * MTILES * NB / 8, 256, 0, stream>>>(sr, si, w1, b1, w2, b2);
  // 5) inverse FFT over H
  fft_cols<1><<<BB * CC * 17, 256, 0, stream>>>(sr, si);
  // 6) irfft over W + AFNO internal residual (h1 += irfft)
  irfft_rows<<<BB * CC * HH / 2, 256, 0, stream>>>(sr, si, h1);
  // 7) out = p2_w @ h1 + p2_b + x
  conv1x1_wmma<1><<<16384, 256, 0, stream>>>(h1, p2w, p2b, x, out);
}